// batch_attention_63178968924388
// MI455X (gfx1250) — compile-verified
//
#include <hip/hip_runtime.h>
#include <math.h>

typedef __attribute__((ext_vector_type(2))) float v2f;
typedef __attribute__((ext_vector_type(8))) float v8f;

#define SDIM   512
#define PLANE  (SDIM * SDIM)      // 262144 floats per mean plane
#define PLANE4 (PLANE / 4)        // 65536 float4 per plane
#define CH     256
#define NV     1024               // length of a / bv

// workspace layout (float offsets)
#define OFF_MEAN   0                       // 2 * PLANE
#define OFF_P      (2 * PLANE)             // 4 * 512 (p111,p112,p221,p222)
#define OFF_A      (OFF_P + 2048)          // 1024
#define OFF_BV     (OFF_A + NV)            // 1024
#define OFF_RMAX   (OFF_BV + NV)           // 1024
#define OFF_RSUM   (OFF_RMAX + NV)         // 1024
#define OFF_DBV    (OFF_RSUM + NV)         // 1024
#define OFF_ANEW   (OFF_DBV + NV)          // 1024
#define OFF_W      (OFF_ANEW + NV)         // 1024
#define OFF_BNEW   (OFF_W + NV)            // 1024
#define OFF_SIG    (OFF_BNEW + NV)         // 2048 (srow1,scol1,srow2,scol2)

// ---------------------------------------------------------------------------
// 1) channel mean: m[b][i][j] = (1/256) sum_c x[b][c][i][j]. float4 streaming.
// 131072 threads, each owns one float4 pixel group, walks 256 channels.
// ---------------------------------------------------------------------------
__global__ __launch_bounds__(256) void mean_kernel(const float* __restrict__ x,
                                                   float* __restrict__ mean) {
  const int t   = blockIdx.x * 256 + threadIdx.x;   // 0 .. 131071
  const int b   = t >> 16;                          // PLANE4 groups per batch
  const int rem = t & (PLANE4 - 1);
  const float4* px = (const float4*)x + (size_t)b * (size_t)CH * PLANE4 + rem;
  float ax = 0.f, ay = 0.f, az = 0.f, aw = 0.f;
  for (int c = 0; c < CH; c += 2) {
    if (c + 4 < CH)
      __builtin_prefetch((const void*)(px + (size_t)(c + 4) * PLANE4), 0, 1);
    float4 v0 = px[(size_t)c * PLANE4];
    float4 v1 = px[(size_t)(c + 1) * PLANE4];
    ax += v0.x + v1.x; ay += v0.y + v1.y;
    az += v0.z + v1.z; aw += v0.w + v1.w;
  }
  const float inv = 1.0f / (float)CH;
  float4 o; o.x = ax * inv; o.y = ay * inv; o.z = az * inv; o.w = aw * inv;
  ((float4*)mean)[b * PLANE4 + rem] = o;
}

// ---------------------------------------------------------------------------
// 2) row/col max of the two mean planes. 2048 threads.
//    grp 0: p111 (rowmax m1), 1: p112 (colmax m1), 2: p221, 3: p222
// ---------------------------------------------------------------------------
__global__ __launch_bounds__(256) void rowcolmax_kernel(const float* __restrict__ mean,
                                                        float* __restrict__ P) {
  const int t   = blockIdx.x * 256 + threadIdx.x;   // 0..2047
  const int grp = t >> 9;
  const int idx = t & (SDIM - 1);
  const float* base = mean + (grp >> 1) * PLANE;
  float mx = -__builtin_inff();
  if ((grp & 1) == 0) {                 // row max
    const float* r = base + idx * SDIM;
    for (int j = 0; j < SDIM; ++j) mx = fmaxf(mx, r[j]);
  } else {                              // col max
    const float* c0 = base + idx;
    for (int i = 0; i < SDIM; ++i) mx = fmaxf(mx, c0[i * SDIM]);
  }
  P[t] = mx;
}

// ---------------------------------------------------------------------------
// 3) conv3 (zero-padded) on the four 512-vectors, assemble a and bv.
// ---------------------------------------------------------------------------
__global__ __launch_bounds__(256) void conv_kernel(const float* __restrict__ P,
    const float* __restrict__ w11, const float* __restrict__ b11,
    const float* __restrict__ w12, const float* __restrict__ b12,
    const float* __restrict__ w21, const float* __restrict__ b21,
    const float* __restrict__ w22, const float* __restrict__ b22,
    float* __restrict__ A, float* __restrict__ BV) {
  const int t   = blockIdx.x * 256 + threadIdx.x;   // 0..2047
  const int grp = t >> 9;
  const int pos = t & (SDIM - 1);
  const float* src = P + grp * SDIM;
  const float* w; const float* bb;
  if      (grp == 0) { w = w11; bb = b11; }
  else if (grp == 1) { w = w12; bb = b12; }
  else if (grp == 2) { w = w21; bb = b21; }
  else               { w = w22; bb = b22; }
  const float left  = (pos > 0)        ? src[pos - 1] : 0.0f;
  const float mid   = src[pos];
  const float right = (pos < SDIM - 1) ? src[pos + 1] : 0.0f;
  const float r = w[0] * left + w[1] * mid + w[2] * right + bb[0];
  float* dst = (grp < 2) ? A : BV;
  dst[(grp & 1) * SDIM + pos] = r;
}

// ---------------------------------------------------------------------------
// 4) per-row max of z_ij = |a_i - bv_j| (softmax stabilizer). 1024 threads.
// ---------------------------------------------------------------------------
__global__ __launch_bounds__(256) void rowmax_kernel(const float* __restrict__ A,
                                                     const float* __restrict__ BV,
                                                     float* __restrict__ RMAX) {
  const int i = blockIdx.x * 256 + threadIdx.x;     // 0..1023
  const float ai = A[i];
  float mx = -__builtin_inff();
  for (int j = 0; j < NV; ++j) mx = fmaxf(mx, fabsf(ai - BV[j]));
  RMAX[i] = mx;
}

// ---------------------------------------------------------------------------
// 5) WMMA row pass: D(1024x16) = E(1024x1024) x [1 | bv | 0...]
//    wave w owns rows 16w..16w+15; K-loop in steps of 4 using
//    V_WMMA_F32_16X16X4_F32.  A layout (16x4 f32, ISA 7.12.2):
//      lane<16: rows, VGPR0=K0, VGPR1=K1 ; lane>=16: same rows, K2/K3.
//    C layout: vgpr v, lane l -> (M = v + 8*(l>>4), N = l&15).
//    Result: D[i,0] = rowsum_i, D[i,1] = (E@bv)_i.
// ---------------------------------------------------------------------------
__global__ __launch_bounds__(256) void attn_rows_wmma(const float* __restrict__ A,
    const float* __restrict__ BV, const float* __restrict__ RMAX,
    float* __restrict__ RSUM, float* __restrict__ DBV) {
  const int lane = threadIdx.x & 31;
  const int wave = blockIdx.x * 8 + (threadIdx.x >> 5);   // 0..63
  const int sub  = lane & 15;
  const int hi   = lane >> 4;
  const int r0   = wave * 16;
  const float ai = A[r0 + sub];
  const float mi = RMAX[r0 + sub];
  v8f c = {};
  for (int k = 0; k < NV; k += 4) {
    const int kk   = k + 2 * hi;
    const float b0 = BV[kk];
    const float b1 = BV[kk + 1];
    v2f av, bm;
    av.x = expf(fabsf(ai - b0) - mi);            // E[row, kk]
    av.y = expf(fabsf(ai - b1) - mi);            // E[row, kk+1]
    bm.x = (sub == 0) ? 1.0f : ((sub == 1) ? b0 : 0.0f);
    bm.y = (sub == 0) ? 1.0f : ((sub == 1) ? b1 : 0.0f);
    c = __builtin_amdgcn_wmma_f32_16x16x4_f32(false, av, false, bm,
                                              (short)0, c, false, false);
  }
#pragma unroll
  for (int v = 0; v < 8; ++v) {
    const int m = r0 + v + 8 * hi;
    if (sub == 0)      RSUM[m] = c[v];
    else if (sub == 1) DBV[m]  = c[v];
  }
}

// ---------------------------------------------------------------------------
// 6) a_new_i = dotbv_i / s_i + a_i ;  w_i = a_i / s_i
// ---------------------------------------------------------------------------
__global__ __launch_bounds__(256) void finish_rows_kernel(const float* __restrict__ A,
    const float* __restrict__ RSUM, const float* __restrict__ DBV,
    float* __restrict__ ANEW, float* __restrict__ W) {
  const int i = blockIdx.x * 256 + threadIdx.x;   // 0..1023
  const float inv = 1.0f / RSUM[i];
  ANEW[i] = DBV[i] * inv + A[i];
  W[i]    = A[i] * inv;
}

// ---------------------------------------------------------------------------
// 7) WMMA col pass: b_new_j = sum_i w_i * e_ij + bv_j  via D = E^T x [w | 0...]
//    wave w owns columns 16w..16w+15 (as A-matrix rows); K runs over i.
// ---------------------------------------------------------------------------
__global__ __launch_bounds__(256) void attn_cols_wmma(const float* __restrict__ A,
    const float* __restrict__ BV, const float* __restrict__ RMAX,
    const float* __restrict__ W, float* __restrict__ BNEW) {
  const int lane = threadIdx.x & 31;
  const int wave = blockIdx.x * 8 + (threadIdx.x >> 5);   // 0..63
  const int sub  = lane & 15;
  const int hi   = lane >> 4;
  const int j0   = wave * 16;
  const float bj = BV[j0 + sub];
  v8f c = {};
  for (int k = 0; k < NV; k += 4) {
    const int kk   = k + 2 * hi;
    const float a0 = A[kk],     a1 = A[kk + 1];
    const float m0 = RMAX[kk],  m1 = RMAX[kk + 1];
    const float w0 = W[kk],     w1 = W[kk + 1];
    v2f av, bm;
    av.x = expf(fabsf(a0 - bj) - m0);            // E^T[j, kk]   = E[kk, j]
    av.y = expf(fabsf(a1 - bj) - m1);            // E^T[j, kk+1]
    bm.x = (sub == 0) ? w0 : 0.0f;
    bm.y = (sub == 0) ? w1 : 0.0f;
    c = __builtin_amdgcn_wmma_f32_16x16x4_f32(false, av, false, bm,
                                              (short)0, c, false, false);
  }
#pragma unroll
  for (int v = 0; v < 8; ++v) {
    if (sub == 0) {
      const int jj = j0 + v + 8 * hi;
      BNEW[jj] = c[v] + BV[jj];
    }
  }
}

// ---------------------------------------------------------------------------
// 8) sigmoids: SIG[0:1024] = sig(a_new), SIG[1024:2048] = sig(b_new)
//    (layout matches srow1,scol1,srow2,scol2)
// ---------------------------------------------------------------------------
__global__ __launch_bounds__(256) void sig_kernel(const float* __restrict__ ANEW,
                                                  const float* __restrict__ BNEW,
                                                  float* __restrict__ SIG) {
  const int t = blockIdx.x * 256 + threadIdx.x;   // 0..2047
  const float v = (t < NV) ? ANEW[t] : BNEW[t - NV];
  SIG[t] = 1.0f / (1.0f + expf(-v));
}

// ---------------------------------------------------------------------------
// 9) fused gate: out[b,c,i,j] = x[b,c,i,j] * m_b[i,j] * srow_b[i] * scol_b[j]
//    float4 streaming; mean + sig live in L2.
// ---------------------------------------------------------------------------
__global__ __launch_bounds__(256) void gate_kernel(const float* __restrict__ x,
    const float* __restrict__ mean, const float* __restrict__ sig,
    float* __restrict__ out) {
  const size_t t  = (size_t)blockIdx.x * 256 + threadIdx.x;  // 0 .. 33554431
  const int b     = (int)(t >> 24);                          // CH*PLANE4 per batch
  const int rem   = (int)(t & (PLANE4 - 1));                 // pixel float4 index
  const int i     = rem >> 7;                                // row
  const int jv    = rem & 127;                               // float4 col group
  const float4 xv = ((const float4*)x)[t];
  const float4 mv = ((const float4*)mean)[b * PLANE4 + rem];
  const float* sb = sig + b * NV;
  const float  sr = sb[i];
  const float4 sc = ((const float4*)(sb + SDIM))[jv];
  float4 o;
  o.x = xv.x * mv.x * sr * sc.x;
  o.y = xv.y * mv.y * sr * sc.y;
  o.z = xv.z * mv.z * sr * sc.z;
  o.w = xv.w * mv.w * sr * sc.w;
  ((float4*)out)[t] = o;
}

// ---------------------------------------------------------------------------
extern "C" void kernel_launch(void* const* d_in, const int* in_sizes, int n_in,
                              void* d_out, int out_size, void* d_ws, size_t ws_size,
                              hipStream_t stream) {
  (void)in_sizes; (void)n_in; (void)out_size; (void)ws_size;
  const float* x   = (const float*)d_in[0];
  const float* w11 = (const float*)d_in[1];
  const float* b11 = (const float*)d_in[2];
  const float* w12 = (const float*)d_in[3];
  const float* b12 = (const float*)d_in[4];
  const float* w21 = (const float*)d_in[5];
  const float* b21 = (const float*)d_in[6];
  const float* w22 = (const float*)d_in[7];
  const float* b22 = (const float*)d_in[8];
  float* out = (float*)d_out;
  float* ws  = (float*)d_ws;

  float* MEAN = ws + OFF_MEAN;
  float* P    = ws + OFF_P;
  float* A    = ws + OFF_A;
  float* BV   = ws + OFF_BV;
  float* RMAX = ws + OFF_RMAX;
  float* RSUM = ws + OFF_RSUM;
  float* DBV  = ws + OFF_DBV;
  float* ANEW = ws + OFF_ANEW;
  float* W    = ws + OFF_W;
  float* BNEW = ws + OFF_BNEW;
  float* SIG  = ws + OFF_SIG;

  // 1) channel means (reads 512 MB)
  mean_kernel<<<512, 256, 0, stream>>>(x, MEAN);
  // 2) row/col maxima of the two mean planes
  rowcolmax_kernel<<<8, 256, 0, stream>>>(MEAN, P);
  // 3) conv3 -> a, bv
  conv_kernel<<<8, 256, 0, stream>>>(P, w11, b11, w12, b12, w21, b21, w22, b22,
                                     A, BV);
  // 4) per-row max for stable softmax
  rowmax_kernel<<<4, 256, 0, stream>>>(A, BV, RMAX);
  // 5) WMMA row pass: rowsum + E@bv
  attn_rows_wmma<<<8, 256, 0, stream>>>(A, BV, RMAX, RSUM, DBV);
  // 6) a_new, w
  finish_rows_kernel<<<4, 256, 0, stream>>>(A, RSUM, DBV, ANEW, W);
  // 7) WMMA col pass: b_new
  attn_cols_wmma<<<8, 256, 0, stream>>>(A, BV, RMAX, W, BNEW);
  // 8) sigmoids
  sig_kernel<<<8, 256, 0, stream>>>(ANEW, BNEW, SIG);
  // 9) fused gate (reads 512 MB, writes 512 MB)
  gate_kernel<<<131072, 256, 0, stream>>>(x, MEAN, SIG, out);
}